// Attention_67156108640667
// MI455X (gfx1250) — compile-verified
//
#include <hip/hip_runtime.h>
#include <math.h>

typedef __attribute__((ext_vector_type(2))) float v2f;
typedef __attribute__((ext_vector_type(8))) float v8f;

#define B_ 64
#define T_ 4096
#define D_ 512

// ---------------------------------------------------------------------------
// Pass 1: energies[r] = dot(E[r, :], v)  for r in [0, B*T)   (flat rows)
// One wave computes a 16-row tile with V_WMMA_F32_16X16X4_F32 chained over K.
// B-matrix: v chunk replicated into all 16 columns -> every accumulator
// column equals the 16 dot products.
// A 16x4 layout: lanes 0-15 hold K=0(v0),K=1(v1); lanes 16-31 hold K=2,K=3.
// ---------------------------------------------------------------------------
__global__ __launch_bounds__(256) void scores_wmma(const float* __restrict__ E,
                                                   const float* __restrict__ v,
                                                   float* __restrict__ energies) {
  const int lane = threadIdx.x & 31;
  const int wave = threadIdx.x >> 5;
  const long rowBase = (long)blockIdx.x * 128 + (long)wave * 16;
  const int m  = lane & 15;         // A row for this lane
  const int kh = (lane >> 4) * 2;   // K sub-offset for this half-wave

  const float* __restrict__ rowPtr = E + (rowBase + m) * (long)D_;
  v8f c = {};
  for (int k0 = 0; k0 < D_; k0 += 4) {
    v2f a, b;
    a.x = rowPtr[k0 + kh];
    a.y = rowPtr[k0 + kh + 1];
    b.x = v[k0 + kh];          // B[k, n] = v[k] for all n
    b.y = v[k0 + kh + 1];
    c = __builtin_amdgcn_wmma_f32_16x16x4_f32(false, a, false, b,
                                              (short)0, c, false, false);
  }
  // C layout: VGPR i, lanes 0-15 -> M=i, lanes 16-31 -> M=i+8 (N = lane&15;
  // all N identical). Lane 0 writes rows 0..7, lane 16 writes rows 8..15.
  if ((lane & 15) == 0) {
    const long out0 = rowBase + (long)(lane >> 4) * 8;
#pragma unroll
    for (int i = 0; i < 8; ++i) energies[out0 + i] = c[i];
  }
}

// ---------------------------------------------------------------------------
// Pass 2: row softmax over T per batch. Tiny traffic (1 MB); plain LDS tree.
// ---------------------------------------------------------------------------
__global__ __launch_bounds__(256) void softmax_rows(const float* __restrict__ energies,
                                                    float* __restrict__ weights) {
  __shared__ float red[256];
  const int b = blockIdx.x;
  const float* __restrict__ e = energies + (long)b * T_;
  float* __restrict__ w = weights + (long)b * T_;

  float m = -INFINITY;
  for (int t = threadIdx.x; t < T_; t += 256) m = fmaxf(m, e[t]);
  red[threadIdx.x] = m;
  __syncthreads();
  for (int s = 128; s > 0; s >>= 1) {
    if (threadIdx.x < s)
      red[threadIdx.x] = fmaxf(red[threadIdx.x], red[threadIdx.x + s]);
    __syncthreads();
  }
  m = red[0];
  __syncthreads();

  float sum = 0.f;
  for (int t = threadIdx.x; t < T_; t += 256) sum += __expf(e[t] - m);
  red[threadIdx.x] = sum;
  __syncthreads();
  for (int s = 128; s > 0; s >>= 1) {
    if (threadIdx.x < s) red[threadIdx.x] += red[threadIdx.x + s];
    __syncthreads();
  }
  const float inv = 1.0f / red[0];
  for (int t = threadIdx.x; t < T_; t += 256) w[t] = __expf(e[t] - m) * inv;
}

// ---------------------------------------------------------------------------
// Pass 3: context[b, d] = sum_t w[b,t] * E[b,t,d].
// Block = (d-tile of 16, batch). 8 waves split T; each wave chains WMMAs with
// A = weights chunk replicated over rows -> every accumulator row holds the
// 16-wide context partial. Partials combined through LDS.
// ---------------------------------------------------------------------------
__global__ __launch_bounds__(256) void context_wmma(const float* __restrict__ E,
                                                    const float* __restrict__ weights,
                                                    float* __restrict__ context) {
  __shared__ float part[8][16];
  const int lane = threadIdx.x & 31;
  const int wave = threadIdx.x >> 5;
  const int d0 = blockIdx.x * 16;
  const int b  = blockIdx.y;
  const int n  = lane & 15;
  const int kh = (lane >> 4) * 2;

  const float* __restrict__ w  = weights + (long)b * T_;
  const float* __restrict__ Eb = E + (long)b * T_ * D_ + d0 + n;

  v8f c = {};
  for (int t0 = wave * 4; t0 < T_; t0 += 32) {
    v2f a, bm;
    a.x  = w[t0 + kh];                       // A[m,k] = w[t0+k], all rows m
    a.y  = w[t0 + kh + 1];
    bm.x = Eb[(long)(t0 + kh) * D_];         // B[k,n] = E[t0+k, d0+n]
    bm.y = Eb[(long)(t0 + kh + 1) * D_];
    __builtin_prefetch(&Eb[(long)(t0 + 32 + kh) * D_], 0, 1);
    c = __builtin_amdgcn_wmma_f32_16x16x4_f32(false, a, false, bm,
                                              (short)0, c, false, false);
  }
  // Row M=0 of C (c[0], lanes 0-15) holds this wave's context partial.
  if (lane < 16) part[wave][lane] = c[0];
  __syncthreads();
  if (threadIdx.x < 16) {
    float s = 0.f;
#pragma unroll
    for (int wv = 0; wv < 8; ++wv) s += part[wv][threadIdx.x];
    context[(long)b * D_ + d0 + threadIdx.x] = s;
  }
}

// ---------------------------------------------------------------------------
// d_in[0] = encoder_outputs [B,T,D] f32, d_in[1] = attn_weights_param [D,1]
// d_out   = context [B,D] f32 then weights [B,T] f32 (tuple order, flat)
// d_ws    = energies [B,T] f32 scratch (1 MiB)
// ---------------------------------------------------------------------------
extern "C" void kernel_launch(void* const* d_in, const int* in_sizes, int n_in,
                              void* d_out, int out_size, void* d_ws, size_t ws_size,
                              hipStream_t stream) {
  const float* E = (const float*)d_in[0];
  const float* v = (const float*)d_in[1];
  float* context  = (float*)d_out;                   // B*D elements
  float* weights  = (float*)d_out + (long)B_ * D_;   // B*T elements
  float* energies = (float*)d_ws;                    // B*T elements

  scores_wmma<<<(B_ * T_) / 128, 256, 0, stream>>>(E, v, energies);
  softmax_rows<<<B_, 256, 0, stream>>>(energies, weights);
  context_wmma<<<dim3(D_ / 16, B_), 256, 0, stream>>>(E, weights, context);
}